// PagedAttention_10247791968906
// MI455X (gfx1250) — compile-verified
//
#include <hip/hip_runtime.h>
#include <stdint.h>

typedef float v2f __attribute__((ext_vector_type(2)));
typedef float v4f __attribute__((ext_vector_type(4)));
typedef float v8f __attribute__((ext_vector_type(8)));
typedef unsigned int v4u __attribute__((ext_vector_type(4)));
typedef int v4i __attribute__((ext_vector_type(4)));
typedef int v8i __attribute__((ext_vector_type(8)));

#if __has_builtin(__builtin_amdgcn_tensor_load_to_lds)
#define HAVE_TDM 1
#else
#define HAVE_TDM 0
#endif

namespace {
constexpr int kB = 32, kH = 32, kHkv = 8, kD = 128, kG = 4, kBS = 16, kMaxBlk = 128;
constexpr int ROW_DW   = 132;                 // 128 payload dwords + 4 pad dwords (bank decorrelation)
constexpr int TILE_DW  = 16 * ROW_DW;         // one 16x128 f32 tile in LDS
constexpr int TILES_OFF = TILE_DW;            // Q tile occupies [0, TILE_DW)
constexpr int MACC_OFF = TILES_OFF + 16 * TILE_DW;   // 4 waves * 2 bufs * {K,V}
constexpr int MST_OFF  = MACC_OFF + 4 * kG * kD;
constexpr int SMEM_DW  = MST_OFF + 4 * kG * 2;
}

// ---------------- RoPE (positions = context_lens), writes q_rope / k_rope to workspace
__global__ void rope_kernel(const float* __restrict__ q, const float* __restrict__ k,
                            const int* __restrict__ ctx,
                            float* __restrict__ qo, float* __restrict__ ko) {
  int idx = blockIdx.x * blockDim.x + threadIdx.x;
  int total = kB * (kH + kHkv) * kD;
  if (idx >= total) return;
  int d = idx % kD;
  int row = idx / kD;
  int b = row / (kH + kHkv);
  int h = row % (kH + kHkv);
  const float* src;
  float* dst;
  if (h < kH) { src = q + ((size_t)b * kH + h) * kD;          dst = qo + ((size_t)b * kH + h) * kD; }
  else        { src = k + ((size_t)b * kHkv + (h - kH)) * kD; dst = ko + ((size_t)b * kHkv + (h - kH)) * kD; }
  float pos = (float)ctx[b];
  int i = d & 63;                                   // d % (D/2)
  float inv = __expf(-(float)i * (9.2103403719761836f / 64.0f)); // 10000^(-i/64)
  float ang = pos * inv;
  float c = __cosf(ang), s = __sinf(ang);
  int dr = (d < 64) ? d + 64 : d - 64;
  float negm = (dr & 1) ? 1.0f : -1.0f;             // neg_mask = [-1,1] tiled
  dst[d] = src[d] * c + src[dr] * negm * s;
}

// ---------------- tile movers: TDM (async, padded rows) with sync fallback
__device__ __forceinline__ void tile_load(float* sm, unsigned dst_dw, const float* gsrc, int lane) {
#if HAVE_TDM
  (void)lane;
  unsigned long long ga = (unsigned long long)(uintptr_t)gsrc;
  unsigned lds_byte = (unsigned)(uintptr_t)(&sm[dst_dw]);
  v4u g0;
  g0[0] = 1u;                                              // count = 1 (valid descriptor)
  g0[1] = lds_byte;                                        // LDS byte address
  g0[2] = (unsigned)(ga & 0xffffffffull);                  // global addr [31:0]
  g0[3] = (unsigned)((ga >> 32) & 0x1ffffffull) | 0x80000000u; // addr [56:32] | type=2
  v8i g1;
  g1[0] = (2 << 16) | (1 << 20) | (6 << 22) | (3 << 25);   // 4B elems, pad_en, 4dw pad / 128dw
  g1[1] = (128 << 16);                                     // tensor_dim0 = 128
  g1[2] = (16 << 16);                                      // tensor_dim1 = 16
  g1[3] = (128 << 16);                                     // tile_dim0 = 128
  g1[4] = 16;                                              // tile_dim1 = 16, tile_dim2 = 0
  g1[5] = kHkv * kD;                                       // tensor_dim0_stride = 1024 floats
  g1[6] = 0;
  g1[7] = 0;
  v4i z4 = {0, 0, 0, 0};
  v8i z8 = {0, 0, 0, 0, 0, 0, 0, 0};
  __builtin_amdgcn_tensor_load_to_lds(g0, g1, z4, z4, z8, 0);
#else
  for (int i = lane; i < 512; i += 32) {                   // 16 rows x 32 v4f
    int r = i >> 5, q = i & 31;
    v4f v = *(const v4f*)(gsrc + (size_t)r * (kHkv * kD) + q * 4);
    *(v4f*)(&sm[dst_dw + (unsigned)(r * ROW_DW + q * 4)]) = v;
  }
#endif
}

template <int N>
__device__ __forceinline__ void tile_wait() {
#if HAVE_TDM
  __builtin_amdgcn_s_wait_tensorcnt(N);   // constant integer required
#endif
}

// ---------------- flash-decoding paged attention, WMMA f32 16x16x4
__global__ void __launch_bounds__(128)
paged_attn_kernel(const float* __restrict__ qr, const float* __restrict__ kr,
                  const float* __restrict__ vnew, const float* __restrict__ kc,
                  const float* __restrict__ vc, const int* __restrict__ btab,
                  const int* __restrict__ ctx, float* __restrict__ out) {
  extern __shared__ float sm[];
  const int tid  = threadIdx.x;
  const int wave = tid >> 5;
  const int lane = tid & 31;
  const int hi   = lane >> 4;
  const int lm   = lane & 15;
  const int h = blockIdx.x;   // kv head
  const int b = blockIdx.y;   // sequence

  const int ctxlen = ctx[b];
  const int nchunks = (ctxlen + kBS - 1) / kBS;
  const int tstar = ctxlen - 1;
  const int cstar = tstar >> 4;
  const int pstar = tstar & 15;

  // stage scaled Q (rows 0..3 = real GQA rows, rows 4..15 zero)
  const float qscale = 0.0883883476483184f; // 128^-0.5
  for (int i = tid; i < 16 * 128; i += 128) {
    int r = i >> 7, d = i & 127;
    float v = 0.0f;
    if (r < kG) v = qr[((size_t)b * kH + h * kG + r) * kD + d] * qscale;
    sm[r * ROW_DW + d] = v;
  }
  __syncthreads();

  // hoist this lane's Q slice (B-matrix operand, loop-invariant) into registers:
  // step j needs Q[row=lm, d=4j+2hi .. +1]  -> 32 x v2f = 64 VGPRs
  const int dbase = 2 * hi;
  v2f qv[32];
#pragma unroll
  for (int j = 0; j < 32; ++j)
    qv[j] = *(const v2f*)(&sm[(unsigned)(lm * ROW_DW + 4 * j + dbase)]);

  auto issue = [&](int c, int pp) {
    int blk = __builtin_amdgcn_readfirstlane(btab[b * kMaxBlk + c]);
    size_t base = ((size_t)blk * kBS * kHkv + h) * kD;
    unsigned kb = (unsigned)(TILES_OFF + (wave * 4 + pp * 2) * TILE_DW);
    tile_load(sm, kb,           kc + base, lane);
    tile_load(sm, kb + TILE_DW, vc + base, lane);
  };

  float mrow = -3.0e38f, lrow = 0.0f;
  v8f acc[8] = {};

  if (wave < nchunks) issue(wave, 0);
  int pp = 0;
  for (int c = wave; c < nchunks; c += 4) {
    int nx = c + 4;
    if (nx < nchunks) { issue(nx, pp ^ 1); tile_wait<2>(); }
    else              { tile_wait<0>(); }

    unsigned kb = (unsigned)(TILES_OFF + (wave * 4 + pp * 2) * TILE_DW);
    unsigned vb = kb + TILE_DW;

    if (c == cstar) {  // patch in RoPE'd new k and new v at the token being written
      const v4f* gk = (const v4f*)(kr   + ((size_t)b * kHkv + h) * kD);
      const v4f* gv = (const v4f*)(vnew + ((size_t)b * kHkv + h) * kD);
      *(v4f*)(&sm[kb + (unsigned)(pstar * ROW_DW + lane * 4)]) = gk[lane];
      *(v4f*)(&sm[vb + (unsigned)(pstar * ROW_DW + lane * 4)]) = gv[lane];
    }

    // ---- S^T = K(16tok x 128d) * Q^T : 32 wmma steps over d (Q operand from registers)
    v8f s = {};
#pragma unroll
    for (int j = 0; j < 32; ++j) {
      v2f a = *(const v2f*)(&sm[kb + (unsigned)(lm * ROW_DW + 4 * j + dbase)]); // K[tok=lm, d..]
      s = __builtin_amdgcn_wmma_f32_16x16x4_f32(false, a, false, qv[j], (short)0, s, false, false);
    }

    // lane(lm,hi) vgpr g holds S[qrow=lm, token=g+8*hi]; mask invalid tokens
    int tb = c * kBS + 8 * hi;
    float p8[8];
    float cmax = -3.0e38f;
#pragma unroll
    for (int g = 0; g < 8; ++g) {
      float sv = (tb + g < ctxlen) ? s[g] : -3.0e38f;
      s[g] = sv;
      cmax = fmaxf(cmax, sv);
    }
    cmax = fmaxf(cmax, __shfl_xor(cmax, 16, 32));
    float mnew = fmaxf(mrow, cmax);
    float psum = 0.0f;
#pragma unroll
    for (int g = 0; g < 8; ++g) { p8[g] = __expf(s[g] - mnew); psum += p8[g]; }
    psum += __shfl_xor(psum, 16, 32);
    float alpha = __expf(mrow - mnew);
    lrow = lrow * alpha + psum;
    mrow = mnew;

    // rescale accumulators (acc qrow = g + 8*hi)
#pragma unroll
    for (int g = 0; g < 8; ++g) {
      float af = __shfl(alpha, g + 8 * hi, 32);
#pragma unroll
      for (int dt = 0; dt < 8; ++dt) acc[dt][g] *= af;
    }

    // ---- out += P(16q x 16tok) * V(16tok x 128d) : 4 token-steps x 8 d-tiles
    float x16[8];
#pragma unroll
    for (int g = 0; g < 8; ++g) x16[g] = __shfl_xor(p8[g], 16, 32);

#pragma unroll
    for (int j = 0; j < 4; ++j) {
      v2f a;
      if (j == 0) { a.x = hi ? x16[2] : p8[0]; a.y = hi ? x16[3] : p8[1]; }
      if (j == 1) { a.x = hi ? x16[6] : p8[4]; a.y = hi ? x16[7] : p8[5]; }
      if (j == 2) { a.x = hi ? p8[2] : x16[0]; a.y = hi ? p8[3] : x16[1]; }
      if (j == 3) { a.x = hi ? p8[6] : x16[4]; a.y = hi ? p8[7] : x16[5]; }
      unsigned r0 = vb + (unsigned)((4 * j + 2 * hi) * ROW_DW + lm);
      unsigned r1 = r0 + ROW_DW;
#pragma unroll
      for (int dt = 0; dt < 8; ++dt) {
        v2f bv;
        bv.x = sm[r0 + dt * 16];
        bv.y = sm[r1 + dt * 16];
        acc[dt] = __builtin_amdgcn_wmma_f32_16x16x4_f32(false, a, false, bv, (short)0, acc[dt],
                                                        false, false);
      }
    }
    pp ^= 1;
  }

  // ---- write per-wave partials (only qrows 0..3, held in vgprs 0..3 of lanes 0..15)
  if (hi == 0) {
    if (lm < kG) {
      sm[MST_OFF + (wave * kG + lm) * 2 + 0] = mrow;
      sm[MST_OFF + (wave * kG + lm) * 2 + 1] = lrow;
    }
#pragma unroll
    for (int g = 0; g < kG; ++g)
#pragma unroll
      for (int dt = 0; dt < 8; ++dt)
        sm[MACC_OFF + (wave * kG + g) * kD + dt * 16 + lm] = acc[dt][g];
  }
  __syncthreads();

  // ---- merge 4 waves (flash-decoding combine) and emit
  for (int idx = tid; idx < kG * kD; idx += 128) {
    int g = idx >> 7, d = idx & 127;
    float mstar = -3.0e38f;
    for (int w = 0; w < 4; ++w) mstar = fmaxf(mstar, sm[MST_OFF + (w * kG + g) * 2]);
    float lsum = 0.0f, vsum = 0.0f;
    for (int w = 0; w < 4; ++w) {
      float e = __expf(sm[MST_OFF + (w * kG + g) * 2] - mstar);
      lsum += e * sm[MST_OFF + (w * kG + g) * 2 + 1];
      vsum += e * sm[MACC_OFF + (w * kG + g) * kD + d];
    }
    out[((size_t)b * kH + h * kG + g) * kD + d] = vsum / lsum;
  }
}

extern "C" void kernel_launch(void* const* d_in, const int* in_sizes, int n_in,
                              void* d_out, int out_size, void* d_ws, size_t ws_size,
                              hipStream_t stream) {
  (void)in_sizes; (void)n_in; (void)out_size; (void)ws_size;
  const float* query      = (const float*)d_in[0];
  const float* key        = (const float*)d_in[1];
  const float* value      = (const float*)d_in[2];
  const float* k_cache    = (const float*)d_in[3];
  const float* v_cache    = (const float*)d_in[4];
  const int*   block_tab  = (const int*)d_in[5];
  const int*   context    = (const int*)d_in[6];
  float* out = (float*)d_out;

  float* q_rope = (float*)d_ws;                       // B*H*D floats
  float* k_rope = q_rope + (size_t)kB * kH * kD;      // B*Hkv*D floats

  int total = kB * (kH + kHkv) * kD;
  rope_kernel<<<(total + 255) / 256, 256, 0, stream>>>(query, key, context, q_rope, k_rope);

  size_t shmem = (size_t)SMEM_DW * sizeof(float);     // ~152 KB (WGP has 320 KB LDS)
  (void)hipFuncSetAttribute((const void*)paged_attn_kernel,
                            hipFuncAttributeMaxDynamicSharedMemorySize, (int)shmem);
  paged_attn_kernel<<<dim3(kHkv, kB), 128, shmem, stream>>>(
      q_rope, k_rope, value, k_cache, v_cache, block_tab, context, out);
}